// ARMA_72241349918728
// MI455X (gfx1250) — compile-verified
//
#include <hip/hip_runtime.h>
#include <math.h>

#define NNODES 100000
#define NEDGES 3200000
#define FIN    100
#define FHID   32
#define FOUT   2
#define NTILES (NNODES / 16)   // 6250, exact

typedef float v2f __attribute__((ext_vector_type(2)));
typedef float v8f __attribute__((ext_vector_type(8)));

// ---------------------------------------------------------------- utilities
__global__ void zero_kernel(float* __restrict__ p, int n) {
    int i = blockIdx.x * blockDim.x + threadIdx.x;
    if (i < n) p[i] = 0.0f;
}

__global__ void deg_kernel(const int* __restrict__ col, float* __restrict__ deg) {
    int e = blockIdx.x * blockDim.x + threadIdx.x;
    if (e < NEDGES) atomicAdd(&deg[col[e]], 1.0f);
}

__global__ void dinv_kernel(float* __restrict__ deg) {
    int n = blockIdx.x * blockDim.x + threadIdx.x;
    if (n < NNODES) {
        float d = deg[n];
        deg[n] = (d > 0.0f) ? rsqrtf(d) : 0.0f;
    }
}

// ------------------------------------------------ layer-1 GEMM via f32 WMMA
// h     = x @ W1_init              [N, 32]
// hroot = x @ W1_root + b1         [N, 32]
// One wave per 16-row tile; 25 K-steps of V_WMMA_F32_16X16X4_F32,
// 4 accumulators (2 col-tiles x 2 weight matrices). W staged in LDS.
__global__ void __launch_bounds__(256)
gemm1_kernel(const float* __restrict__ x,
             const float* __restrict__ W1i,
             const float* __restrict__ W1r,
             const float* __restrict__ b1,
             float* __restrict__ h,
             float* __restrict__ hroot) {
    __shared__ float sWi[FIN * FHID];
    __shared__ float sWr[FIN * FHID];
    __shared__ float sb[FHID];

    for (int i = threadIdx.x; i < FIN * FHID; i += blockDim.x) {
        sWi[i] = W1i[i];
        sWr[i] = W1r[i];
    }
    if (threadIdx.x < FHID) sb[threadIdx.x] = b1[threadIdx.x];
    __syncthreads();

    const int wave = threadIdx.x >> 5;          // 0..7
    const int lane = threadIdx.x & 31;
    const int tile = blockIdx.x * 8 + wave;     // 16-row tile index
    if (tile >= NTILES) return;                 // wave-uniform: EXEC stays full

    const int row0  = tile * 16;
    const int laneM = lane & 15;                // matrix row (A) / col (B,C,D)
    const int koff  = (lane >> 4) * 2;          // 0 for lanes 0-15, 2 for 16-31

    v8f acc_i0 = {}; v8f acc_i1 = {};           // cols 0-15 / 16-31 of x@W_init
    v8f acc_r0 = {}; v8f acc_r1 = {};           // cols 0-15 / 16-31 of x@W_root

    const float* xrow = x + (size_t)(row0 + laneM) * FIN + koff;

    for (int k0 = 0; k0 < FIN; k0 += 4) {
        // A-matrix 16x4 f32: lane holds x[row0+laneM][k0+koff +{0,1}]
        v2f a;
        a.x = xrow[k0];
        a.y = xrow[k0 + 1];

        // B-matrix 4x16 f32 (row-major W[K][32] in LDS):
        // vgpr0 = W[k0+koff][c0+laneM], vgpr1 = W[k0+koff+1][c0+laneM]
        const float* wi = sWi + (k0 + koff) * FHID;
        const float* wr = sWr + (k0 + koff) * FHID;
        v2f bi0, bi1, br0, br1;
        bi0.x = wi[laneM];            bi0.y = wi[FHID + laneM];
        bi1.x = wi[16 + laneM];       bi1.y = wi[FHID + 16 + laneM];
        br0.x = wr[laneM];            br0.y = wr[FHID + laneM];
        br1.x = wr[16 + laneM];       br1.y = wr[FHID + 16 + laneM];

        acc_i0 = __builtin_amdgcn_wmma_f32_16x16x4_f32(false, a, false, bi0,
                                                       (short)0, acc_i0, false, false);
        acc_i1 = __builtin_amdgcn_wmma_f32_16x16x4_f32(false, a, false, bi1,
                                                       (short)0, acc_i1, false, false);
        acc_r0 = __builtin_amdgcn_wmma_f32_16x16x4_f32(false, a, false, br0,
                                                       (short)0, acc_r0, false, false);
        acc_r1 = __builtin_amdgcn_wmma_f32_16x16x4_f32(false, a, false, br1,
                                                       (short)0, acc_r1, false, false);
    }

    // C/D layout: vgpr j -> row = row0 + (lane>=16 ? 8 : 0) + j, col = c0+laneM
    const int mbase = row0 + (lane >> 4) * 8;
    const float bias_lo = sb[laneM];
    const float bias_hi = sb[16 + laneM];
#pragma unroll
    for (int j = 0; j < 8; ++j) {
        const size_t o = (size_t)(mbase + j) * FHID;
        h[o + laneM]          = acc_i0[j];
        h[o + 16 + laneM]     = acc_i1[j];
        hroot[o + laneM]      = acc_r0[j] + bias_lo;
        hroot[o + 16 + laneM] = acc_r1[j] + bias_hi;
    }
}

// ------------------------------------------------ edge scatter, 32 features
// One wave32 per edge: lane = feature. Coalesced 128B gather + 128B atomic add.
__global__ void scatter1_kernel(const int* __restrict__ row,
                                const int* __restrict__ col,
                                const float* __restrict__ dinv,
                                const float* __restrict__ h,
                                float* __restrict__ agg) {
    const long long t = (long long)blockIdx.x * blockDim.x + threadIdx.x;
    const long long e = t >> 5;
    const int f = (int)(t & 31);
    if (e >= NEDGES) return;
    const int r = row[e];
    const int c = col[e];
    const float w = dinv[r] * dinv[c];
    atomicAdd(&agg[(size_t)c * FHID + f], h[(size_t)r * FHID + f] * w);
}

// h1 = relu(agg1 + hroot)  (written into h buffer)
__global__ void relu_add_kernel(const float* __restrict__ agg,
                                const float* __restrict__ hroot,
                                float* __restrict__ h1) {
    int i = blockIdx.x * blockDim.x + threadIdx.x;
    if (i < NNODES * FHID) h1[i] = fmaxf(agg[i] + hroot[i], 0.0f);
}

// ------------------------------------------------ layer-2 GEMM (32 -> 2, scalar)
__global__ void gemm2_kernel(const float* __restrict__ h1,
                             const float* __restrict__ W2i,
                             const float* __restrict__ W2r,
                             const float* __restrict__ b2,
                             float* __restrict__ ti,
                             float* __restrict__ tr) {
    __shared__ float swi[FHID * FOUT];
    __shared__ float swr[FHID * FOUT];
    __shared__ float sb[FOUT];
    if (threadIdx.x < FHID * FOUT) {
        swi[threadIdx.x] = W2i[threadIdx.x];
        swr[threadIdx.x] = W2r[threadIdx.x];
    }
    if (threadIdx.x < FOUT) sb[threadIdx.x] = b2[threadIdx.x];
    __syncthreads();

    const int n = blockIdx.x * blockDim.x + threadIdx.x;
    if (n >= NNODES) return;
    const float* hp = h1 + (size_t)n * FHID;
    float ai0 = 0.0f, ai1 = 0.0f, ar0 = sb[0], ar1 = sb[1];
#pragma unroll
    for (int k = 0; k < FHID; ++k) {
        const float v = hp[k];
        ai0 = fmaf(v, swi[k * 2 + 0], ai0);
        ai1 = fmaf(v, swi[k * 2 + 1], ai1);
        ar0 = fmaf(v, swr[k * 2 + 0], ar0);
        ar1 = fmaf(v, swr[k * 2 + 1], ar1);
    }
    ti[n * 2 + 0] = ai0;  ti[n * 2 + 1] = ai1;
    tr[n * 2 + 0] = ar0;  tr[n * 2 + 1] = ar1;
}

// ------------------------------------------------ edge scatter, 2 features
__global__ void scatter2_kernel(const int* __restrict__ row,
                                const int* __restrict__ col,
                                const float* __restrict__ dinv,
                                const float* __restrict__ ti,
                                float* __restrict__ agg2) {
    const int e = blockIdx.x * blockDim.x + threadIdx.x;
    if (e >= NEDGES) return;
    const int r = row[e];
    const int c = col[e];
    const float w = dinv[r] * dinv[c];
    atomicAdd(&agg2[(size_t)c * 2 + 0], ti[(size_t)r * 2 + 0] * w);
    atomicAdd(&agg2[(size_t)c * 2 + 1], ti[(size_t)r * 2 + 1] * w);
}

// ------------------------------------------------ relu + 2-class log-softmax
__global__ void final_kernel(const float* __restrict__ agg2,
                             const float* __restrict__ tr,
                             float* __restrict__ out) {
    const int n = blockIdx.x * blockDim.x + threadIdx.x;
    if (n >= NNODES) return;
    float z0 = fmaxf(agg2[n * 2 + 0] + tr[n * 2 + 0], 0.0f);
    float z1 = fmaxf(agg2[n * 2 + 1] + tr[n * 2 + 1], 0.0f);
    const float m = fmaxf(z0, z1);
    const float lse = m + logf(expf(z0 - m) + expf(z1 - m));
    out[n * 2 + 0] = z0 - lse;
    out[n * 2 + 1] = z1 - lse;
}

// ---------------------------------------------------------------- launcher
extern "C" void kernel_launch(void* const* d_in, const int* in_sizes, int n_in,
                              void* d_out, int out_size, void* d_ws, size_t ws_size,
                              hipStream_t stream) {
    const float* x   = (const float*)d_in[0];
    const int*   ei  = (const int*)d_in[1];     // [2, E]
    const int*   row = ei;
    const int*   col = ei + NEDGES;
    const float* W1i = (const float*)d_in[2];
    const float* W1r = (const float*)d_in[3];
    const float* b1  = (const float*)d_in[4];
    const float* W2i = (const float*)d_in[5];
    const float* W2r = (const float*)d_in[6];
    const float* b2  = (const float*)d_in[7];
    float* out = (float*)d_out;

    // workspace layout (floats): ~41.2 MB total
    float* ws    = (float*)d_ws;
    float* deg   = ws;                                    // N       (becomes dinv)
    float* h     = deg   + NNODES;                        // N*32    (becomes h1)
    float* hroot = h     + (size_t)NNODES * FHID;         // N*32
    float* agg1  = hroot + (size_t)NNODES * FHID;         // N*32
    float* ti    = agg1  + (size_t)NNODES * FHID;         // N*2
    float* tr    = ti    + (size_t)NNODES * FOUT;         // N*2
    float* agg2  = tr    + (size_t)NNODES * FOUT;         // N*2

    const int B = 256;
    zero_kernel<<<(NNODES + B - 1) / B, B, 0, stream>>>(deg, NNODES);
    zero_kernel<<<(NNODES * FHID + B - 1) / B, B, 0, stream>>>(agg1, NNODES * FHID);
    zero_kernel<<<(NNODES * FOUT + B - 1) / B, B, 0, stream>>>(agg2, NNODES * FOUT);

    deg_kernel<<<(NEDGES + B - 1) / B, B, 0, stream>>>(col, deg);
    dinv_kernel<<<(NNODES + B - 1) / B, B, 0, stream>>>(deg);

    gemm1_kernel<<<(NTILES + 7) / 8, 256, 0, stream>>>(x, W1i, W1r, b1, h, hroot);

    const long long t1 = (long long)NEDGES * 32;
    scatter1_kernel<<<(unsigned)((t1 + B - 1) / B), B, 0, stream>>>(row, col, deg, h, agg1);
    relu_add_kernel<<<(NNODES * FHID + B - 1) / B, B, 0, stream>>>(agg1, hroot, h);

    gemm2_kernel<<<(NNODES + B - 1) / B, B, 0, stream>>>(h, W2i, W2r, b2, ti, tr);
    scatter2_kernel<<<(NEDGES + B - 1) / B, B, 0, stream>>>(row, col, deg, ti, agg2);
    final_kernel<<<(NNODES + B - 1) / B, B, 0, stream>>>(agg2, tr, out);
}